// SumLayer_75239237091660
// MI455X (gfx1250) — compile-verified
//
#include <hip/hip_runtime.h>

// Problem constants (match the reference harness exactly).
constexpr int S_TOT = 4096;   // samples
constexpr int NN    = 2048;   // sum nodes (output cols)
constexpr int NC    = 4096;   // child nodes (ll cols)
constexpr int NNZ_  = 32768;  // nonzeros per sparse matrix
constexpr int TSAMP = 4;      // samples per workgroup
constexpr int THREADS = 512;  // 16 wave32s per workgroup

// ---- CDNA5 async global->LDS path (guarded; falls back to float4 copies) ----
#if defined(__has_builtin)
#  if __has_builtin(__builtin_amdgcn_global_load_async_to_lds_b128) && \
      __has_builtin(__builtin_amdgcn_s_wait_asynccnt)
#    define USE_ASYNC_LDS 1
#  endif
#endif

#ifdef USE_ASYNC_LDS
typedef int v4i __attribute__((ext_vector_type(4)));
typedef __attribute__((address_space(1))) v4i* gbl_v4i_t;
typedef __attribute__((address_space(3))) v4i* lds_v4i_t;
__device__ __forceinline__ void async_copy16(const float* src, float* dst) {
  __builtin_amdgcn_global_load_async_to_lds_b128(
      (gbl_v4i_t)src, (lds_v4i_t)dst, 0, 0);
}
#endif

// ---------------- prep kernels: logz[n] = log(sum exp(w) over row n) --------
__global__ void zero_z_kernel(float* __restrict__ z) {
  int i = blockIdx.x * blockDim.x + threadIdx.x;
  if (i < NN) z[i] = 0.0f;
}

__global__ void zacc_kernel(const float* __restrict__ wd,
                            const int* __restrict__ wr,
                            float* __restrict__ z) {
  int i = blockIdx.x * blockDim.x + threadIdx.x;
  if (i < NNZ_) atomicAdd(&z[wr[i]], __expf(wd[i]));
}

__global__ void log_z_kernel(float* __restrict__ z) {
  int i = blockIdx.x * blockDim.x + threadIdx.x;
  if (i < NN) z[i] = __logf(z[i]);
}

// ---------------- main kernel: 1 workgroup = 4 samples ----------------------
// LDS: lls[8][4096] (ll0/ll1 rows for 4 samples, 128 KB) + acc[4][2048] (32 KB)
// 160 KB dynamic LDS -> 2 workgroups per 320 KB WGP -> 32 wave32s per WGP.
__global__ __launch_bounds__(THREADS)
void sum_layer_main(const float* __restrict__ ll0, const float* __restrict__ ll1,
                    const float* __restrict__ w0d, const int* __restrict__ w0r,
                    const int* __restrict__ w0c,
                    const float* __restrict__ w1d, const int* __restrict__ w1r,
                    const int* __restrict__ w1c,
                    const float* __restrict__ logz, float* __restrict__ out) {
  extern __shared__ float smem[];
  float* lls = smem;            // 8 * NC floats
  float* acc = smem + 8 * NC;   // TSAMP * NN floats

  const int tid = threadIdx.x;
  const int s0  = blockIdx.x * TSAMP;

  // Zero the per-sample accumulators (disjoint LDS region from the staging).
  for (int i = tid; i < TSAMP * NN; i += THREADS) acc[i] = 0.0f;

  // Stage ll rows into LDS. Layout: lls[(t*2 + m) * NC + c].
#ifdef USE_ASYNC_LDS
  for (int j = 0; j < 2 * TSAMP; ++j) {
    const int t = j >> 1, m = j & 1;
    const float* src = (m ? ll1 : ll0) + (size_t)(s0 + t) * NC;
    float* dst = lls + j * NC;
    // 512 lanes * 16B = half a row per call; two calls per row.
    async_copy16(src + tid * 4, dst + tid * 4);
    async_copy16(src + 2048 + tid * 4, dst + 2048 + tid * 4);
  }
  __builtin_amdgcn_s_wait_asynccnt(0);
#else
  for (int j = 0; j < 2 * TSAMP; ++j) {
    const int t = j >> 1, m = j & 1;
    const float4* src =
        (const float4*)((m ? ll1 : ll0) + (size_t)(s0 + t) * NC);
    float4* dst = (float4*)(lls + j * NC);
    for (int q = tid; q < NC / 4; q += THREADS) dst[q] = src[q];
  }
#endif
  __syncthreads();

  // Stream COO entries in b128 chunks (L2-resident); scatter exp(ll + w)
  // into the LDS accumulators via ds_add_f32.
  for (int m = 0; m < 2; ++m) {
    const int4*   wr4 = (const int4*)(m ? w1r : w0r);
    const int4*   wc4 = (const int4*)(m ? w1c : w0c);
    const float4* wd4 = (const float4*)(m ? w1d : w0d);
    const float* lm = lls + m * NC;   // lls[(t*2+m)*NC] = lm + t*2*NC
    for (int k = tid; k < NNZ_ / 4; k += THREADS) {
      const int4   r4 = wr4[k];
      const int4   c4 = wc4[k];
      const float4 w4 = wd4[k];
      const int   rr[4] = {r4.x, r4.y, r4.z, r4.w};
      const int   cc[4] = {c4.x, c4.y, c4.z, c4.w};
      const float ww[4] = {w4.x, w4.y, w4.z, w4.w};
#pragma unroll
      for (int e = 0; e < 4; ++e) {
#pragma unroll
        for (int t = 0; t < TSAMP; ++t) {
          const float v = __expf(lm[t * 2 * NC + cc[e]] + ww[e]);
          atomicAdd(&acc[t * NN + rr[e]], v);
        }
      }
    }
  }
  __syncthreads();

  // out[s, n] = log(acc) - log(z[n]); coalesced non-temporal writes
  // (out is never re-read; keep L2 for the ll / COO streams).
  for (int i = tid; i < TSAMP * NN; i += THREADS) {
    const int t = i >> 11;          // NN == 2048
    const int n = i & (NN - 1);
    const float val = __logf(acc[i]) - logz[n];
    __builtin_nontemporal_store(val, &out[(size_t)(s0 + t) * NN + n]);
  }
}

// ---------------------------------------------------------------------------
extern "C" void kernel_launch(void* const* d_in, const int* in_sizes, int n_in,
                              void* d_out, int out_size, void* d_ws, size_t ws_size,
                              hipStream_t stream) {
  (void)in_sizes; (void)n_in; (void)out_size; (void)ws_size;
  const float* ll0 = (const float*)d_in[0];
  const float* ll1 = (const float*)d_in[1];
  const float* w0d = (const float*)d_in[2];
  const float* w1d = (const float*)d_in[3];
  const int*   w0r = (const int*)d_in[4];
  const int*   w0c = (const int*)d_in[5];
  const int*   w1r = (const int*)d_in[6];
  const int*   w1c = (const int*)d_in[7];
  float* out = (float*)d_out;
  float* z   = (float*)d_ws;  // 2048 floats of scratch -> logz

  zero_z_kernel<<<NN / 256, 256, 0, stream>>>(z);
  zacc_kernel<<<NNZ_ / 256, 256, 0, stream>>>(w0d, w0r, z);
  zacc_kernel<<<NNZ_ / 256, 256, 0, stream>>>(w1d, w1r, z);
  log_z_kernel<<<NN / 256, 256, 0, stream>>>(z);

  const size_t smem_bytes = (size_t)(8 * NC + TSAMP * NN) * sizeof(float); // 160 KB
  sum_layer_main<<<S_TOT / TSAMP, THREADS, smem_bytes, stream>>>(
      ll0, ll1, w0d, w0r, w0c, w1d, w1r, w1c, z, out);
}